// LogTsallisBisect_12421045420948
// MI455X (gfx1250) — compile-verified
//
#include <hip/hip_runtime.h>
#include <math.h>

#define D_CLASSES   32000
#define BLOCK       1024
#define NCHUNK      8                    // 8 float4 chunks per thread
#define VPT         (NCHUNK * 4)         // 32 elements per thread
#define CHUNK_ELEMS (BLOCK * 4)          // 4096 floats per chunk step
#define TAIL_T      832                  // (32000 - 7*4096)/4 active threads in last chunk
#define NWAVES      (BLOCK / 32)
#define SENTINEL    (-3.0e38f)

#if defined(__gfx1250__)
#define ASYNC_STAGE 1
#else
#define ASYNC_STAGE 0
#endif

// ---------------- deterministic wave32 / block reductions ----------------

__device__ __forceinline__ float wave_reduce_sum(float v) {
#pragma unroll
  for (int m = 16; m >= 1; m >>= 1) v += __shfl_xor(v, m, 32);
  return v;
}

__device__ __forceinline__ float wave_reduce_max(float v) {
#pragma unroll
  for (int m = 16; m >= 1; m >>= 1) v = fmaxf(v, __shfl_xor(v, m, 32));
  return v;
}

__device__ __forceinline__ float block_allreduce_sum(float v, float* red) {
  v = wave_reduce_sum(v);
  const int wave = threadIdx.x >> 5;
  const int lane = threadIdx.x & 31;
  __syncthreads();                 // protect red[] reuse from previous call
  if (lane == 0) red[wave] = v;
  __syncthreads();
  if (wave == 0) {
    float t = red[lane];           // NWAVES == 32 == lanes
    t = wave_reduce_sum(t);
    if (lane == 0) red[NWAVES] = t;
  }
  __syncthreads();
  return red[NWAVES];
}

__device__ __forceinline__ float block_allreduce_max(float v, float* red) {
  v = wave_reduce_max(v);
  const int wave = threadIdx.x >> 5;
  const int lane = threadIdx.x & 31;
  __syncthreads();
  if (lane == 0) red[wave] = v;
  __syncthreads();
  if (wave == 0) {
    float t = red[lane];
    t = wave_reduce_max(t);
    if (lane == 0) red[NWAVES] = t;
  }
  __syncthreads();
  return red[NWAVES];
}

// ---------------- kernel: one 1024-thread block (32 waves) per row ----------------

__global__ void __launch_bounds__(BLOCK)
tsallis15_bisect_log_kernel(const float* __restrict__ X, float* __restrict__ Out) {
  __shared__ float red[NWAVES + 1];

  const int tid = threadIdx.x;
  const long long row = blockIdx.x;
  const float* __restrict__ xrow = X + row * (long long)D_CLASSES;
  float* __restrict__ orow = Out + row * (long long)D_CLASSES;

  float xs[VPT];   // register-resident row slice, pre-scaled by (alpha-1)=0.5

#if ASYNC_STAGE
  // ---- Pipelined staging: async global->LDS (double buffer), ds_load to regs ----
  __shared__ __align__(16) float stage[2][CHUNK_ELEMS];   // 2 x 16 KB
  const unsigned lds_off0 = (unsigned)(size_t)(&stage[0][tid * 4]);
  const unsigned lds_off1 = (unsigned)(size_t)(&stage[1][tid * 4]);

#define ISSUE_CHUNK(k)                                                         \
  do {                                                                         \
    if (((k) < NCHUNK - 1) || (tid < TAIL_T)) {                                \
      const float* gsrc_ = xrow + (k) * CHUNK_ELEMS + tid * 4;                 \
      unsigned off_ = ((k) & 1) ? lds_off1 : lds_off0;                         \
      asm volatile("global_load_async_to_lds_b128 %0, %1, off"                 \
                   :: "v"(off_), "v"(gsrc_) : "memory");                       \
    }                                                                          \
  } while (0)

  ISSUE_CHUNK(0);
  ISSUE_CHUNK(1);
#pragma unroll
  for (int k = 0; k < NCHUNK; ++k) {
    // Per-wave async loads complete in order: cnt<=1 means chunk k has landed
    // (chunk k+1 may still be in flight); last chunk waits for everything.
    if (k < NCHUNK - 1) { asm volatile("s_wait_asynccnt 1" ::: "memory"); }
    else                { asm volatile("s_wait_asynccnt 0" ::: "memory"); }
    __syncthreads();                                   // chunk k visible to all waves
    const bool valid = (k < NCHUNK - 1) || (tid < TAIL_T);
    float4 v;
    if (valid) v = *reinterpret_cast<const float4*>(&stage[k & 1][tid * 4]);
    else       v = make_float4(SENTINEL, SENTINEL, SENTINEL, SENTINEL);
    __syncthreads();                                   // all waves done reading buf k&1
    if (k + 2 < NCHUNK) ISSUE_CHUNK(k + 2);            // safe to overwrite buf k&1 now
    xs[k * 4 + 0] = v.x * 0.5f;                        // Xs = X * (alpha - 1)
    xs[k * 4 + 1] = v.y * 0.5f;
    xs[k * 4 + 2] = v.z * 0.5f;
    xs[k * 4 + 3] = v.w * 0.5f;
  }
#undef ISSUE_CHUNK
#else
#pragma unroll
  for (int k = 0; k < NCHUNK; ++k) {
    const bool valid = (k < NCHUNK - 1) || (tid < TAIL_T);
    float4 v;
    if (valid) v = *reinterpret_cast<const float4*>(xrow + k * CHUNK_ELEMS + tid * 4);
    else       v = make_float4(SENTINEL, SENTINEL, SENTINEL, SENTINEL);
    xs[k * 4 + 0] = v.x * 0.5f;
    xs[k * 4 + 1] = v.y * 0.5f;
    xs[k * 4 + 2] = v.z * 0.5f;
    xs[k * 4 + 3] = v.w * 0.5f;
  }
#endif

  // ---- row max ----
  float mx = SENTINEL;
#pragma unroll
  for (int i = 0; i < VPT; ++i) mx = fmaxf(mx, xs[i]);
  const float maxv = block_allreduce_max(mx, red);

  // ---- bisection setup (matches reference exactly) ----
  float tau_lo = maxv - 1.0f;
  const float tau_hi = maxv - 0.005590169943749474f;   // (1/32000)^0.5
  float dm = tau_hi - tau_lo;

  float a0 = 0.f, a1 = 0.f, a2 = 0.f, a3 = 0.f;
#pragma unroll
  for (int i = 0; i < VPT; i += 4) {
    float t0 = fmaxf(xs[i + 0] - tau_lo, 0.f); a0 = fmaf(t0, t0, a0);
    float t1 = fmaxf(xs[i + 1] - tau_lo, 0.f); a1 = fmaf(t1, t1, a1);
    float t2 = fmaxf(xs[i + 2] - tau_lo, 0.f); a2 = fmaf(t2, t2, a2);
    float t3 = fmaxf(xs[i + 3] - tau_lo, 0.f); a3 = fmaf(t3, t3, a3);
  }
  const float f_lo = block_allreduce_sum((a0 + a1) + (a2 + a3), red) - 1.0f;

  // ---- 50 bisection steps ----
  float tau_m = tau_lo;
#pragma unroll 1
  for (int it = 0; it < 50; ++it) {
    dm *= 0.5f;
    tau_m = tau_lo + dm;
    float s0 = 0.f, s1 = 0.f, s2 = 0.f, s3 = 0.f;
#pragma unroll
    for (int i = 0; i < VPT; i += 4) {
      float t0 = fmaxf(xs[i + 0] - tau_m, 0.f); s0 = fmaf(t0, t0, s0);
      float t1 = fmaxf(xs[i + 1] - tau_m, 0.f); s1 = fmaf(t1, t1, s1);
      float t2 = fmaxf(xs[i + 2] - tau_m, 0.f); s2 = fmaf(t2, t2, s2);
      float t3 = fmaxf(xs[i + 3] - tau_m, 0.f); s3 = fmaf(t3, t3, s3);
    }
    const float f_m = block_allreduce_sum((s0 + s1) + (s2 + s3), red) - 1.0f;
    tau_lo = (f_m * f_lo >= 0.f) ? tau_m : tau_lo;     // reference mask rule
  }

  // ---- final p (uses tau_m, as in reference), normalization sum ----
  float sp = 0.f;
#pragma unroll
  for (int i = 0; i < VPT; ++i) {
    float t = fmaxf(xs[i] - tau_m, 0.f);
    float pi = t * t;
    xs[i] = pi;                         // reuse registers: xs now holds p
    sp += pi;
  }
  const float S = block_allreduce_sum(sp, red);
  const float invS = 1.0f / S;

  // ---- write log(p / S), -inf at exact zeros ----
#pragma unroll
  for (int k = 0; k < NCHUNK; ++k) {
    const bool valid = (k < NCHUNK - 1) || (tid < TAIL_T);
    if (!valid) continue;
    float4 o;
    float p0 = xs[k * 4 + 0], p1 = xs[k * 4 + 1], p2 = xs[k * 4 + 2], p3 = xs[k * 4 + 3];
    o.x = (p0 > 0.f) ? logf(p0 * invS) : -__builtin_inff();
    o.y = (p1 > 0.f) ? logf(p1 * invS) : -__builtin_inff();
    o.z = (p2 > 0.f) ? logf(p2 * invS) : -__builtin_inff();
    o.w = (p3 > 0.f) ? logf(p3 * invS) : -__builtin_inff();
    *reinterpret_cast<float4*>(orow + k * CHUNK_ELEMS + tid * 4) = o;
  }
}

extern "C" void kernel_launch(void* const* d_in, const int* in_sizes, int n_in,
                              void* d_out, int out_size, void* d_ws, size_t ws_size,
                              hipStream_t stream) {
  (void)n_in; (void)out_size; (void)d_ws; (void)ws_size;
  const float* X = (const float*)d_in[0];
  float* Out = (float*)d_out;
  const int rows = in_sizes[0] / D_CLASSES;   // 4096
  tsallis15_bisect_log_kernel<<<dim3(rows), dim3(BLOCK), 0, stream>>>(X, Out);
}